// MRF_GCN_27462020891071
// MI455X (gfx1250) — compile-verified
//
#include <hip/hip_runtime.h>

typedef __attribute__((ext_vector_type(2))) float v2f;
typedef __attribute__((ext_vector_type(8))) float v8f;

static constexpr int N_NODES = 50000;
static constexpr int N_EDGES = 1600000;

static inline int cdiv(long a, long b) { return (int)((a + b - 1) / b); }

// ---------------- edge preprocessing ----------------
__global__ void deg_kernel(const int* __restrict__ row, const float* __restrict__ ew,
                           float* __restrict__ deg, int E) {
  int e = blockIdx.x * blockDim.x + threadIdx.x;
  if (e < E) atomicAdd(&deg[row[e]], ew[e]);
}

__global__ void dis_kernel(float* __restrict__ deg, int n) {
  int i = blockIdx.x * blockDim.x + threadIdx.x;
  if (i < n) { float d = deg[i]; deg[i] = d > 0.f ? rsqrtf(d) : 0.f; }
}

__global__ void norm_kernel(const int* __restrict__ row, const int* __restrict__ col,
                            const float* __restrict__ ew, const float* __restrict__ dis,
                            float* __restrict__ norm, int E) {
  int e = blockIdx.x * blockDim.x + threadIdx.x;
  if (e < E) norm[e] = -dis[row[e]] * ew[e] * dis[col[e]];
}

// weight transpose: WT[n*K + k] = W[k*M + n]
__global__ void transpose_w(const float* __restrict__ W, float* __restrict__ WT, int K, int M) {
  int i = blockIdx.x * blockDim.x + threadIdx.x;
  if (i >= K * M) return;
  int n = i / K, k = i - n * K;
  WT[i] = W[k * M + n];
}

// ---------------- sparse propagation: out[col] += norm * h[row] ----------------
__global__ __launch_bounds__(128)
void prop_scatter(const float* __restrict__ h, const int* __restrict__ row,
                  const int* __restrict__ col, const float* __restrict__ norm,
                  float* __restrict__ out, int D) {
  const int e = blockIdx.x;
  const int r = row[e];
  const int c = col[e];
  const float w = norm[e];
  const float4* __restrict__ src = (const float4*)(h + (size_t)r * D);
  float* __restrict__ dst = out + (size_t)c * D;
  const int nchunk = D >> 2;
  for (int i = threadIdx.x; i < nchunk; i += blockDim.x) {
    float4 v = src[i];
    atomicAdd(dst + i * 4 + 0, w * v.x);
    atomicAdd(dst + i * 4 + 1, w * v.y);
    atomicAdd(dst + i * 4 + 2, w * v.z);
    atomicAdd(dst + i * 4 + 3, w * v.w);
  }
}

// T2 = 2*T2 - X  (Chebyshev recurrence)
__global__ void cheb2_combine(float* __restrict__ T2, const float* __restrict__ X, long n) {
  long i = (long)blockIdx.x * blockDim.x + threadIdx.x;
  if (i < n) T2[i] = 2.f * T2[i] - X[i];
}

// ---------------- fused multi-term WMMA GEMM: C = sum_t A_t@B_t^T + bias ----------------
// A_t: [Nrows x K] row-major.  BT_t: [M x K] row-major (pre-transposed weights).
// Wave tile: 64 rows x 32 cols (4x2 f32 accumulators). Block: 8 waves = 256 rows x 64 cols.
// NT/K/M/LDC are compile-time so all strides fold into immediate VMEM offsets.
// Out-of-range rows/cols are CLAMPED for loads (garbage only pollutes never-stored lanes).
template <int NT, int K, int M, int LDC>
__global__ __launch_bounds__(256)
void gemm_cheb(const float* __restrict__ A0, const float* __restrict__ A1,
               const float* __restrict__ A2,
               const float* __restrict__ BT0, const float* __restrict__ BT1,
               const float* __restrict__ BT2,
               int Nrows, const float* __restrict__ bias, float* __restrict__ C)
{
  const int tid  = threadIdx.x;
  const int wave = tid >> 5;
  const int lane = tid & 31;
  const int waveRow = wave & 3;
  const int waveCol = wave >> 2;
  const int rowBase = blockIdx.x * 256 + waveRow * 64;
  const int colBase = blockIdx.y * 64 + waveCol * 32;
  if (rowBase >= Nrows || colBase >= M) return;   // wave-uniform exit

  const int lm = lane & 15;          // M (A) / N (B) index within 16-wide tile
  const int lk = (lane >> 4) << 1;   // K sub-offset within frag: 0 or 2

  bool rowOK[4];
  int  mrow[4];
#pragma unroll
  for (int r = 0; r < 4; ++r) {
    const int m = rowBase + r * 16;
    rowOK[r] = m < Nrows;
    mrow[r]  = min(m + lm, Nrows - 1);
  }
  const int n0v = colBase + lm;
  const int n1v = colBase + 16 + lm;
  const int n0  = min(n0v, M - 1);
  const int n1  = min(n1v, M - 1);

  v8f acc[4][2];
  const v8f vzero = {0.f, 0.f, 0.f, 0.f, 0.f, 0.f, 0.f, 0.f};
#pragma unroll
  for (int r = 0; r < 4; ++r)
#pragma unroll
    for (int c = 0; c < 2; ++c) acc[r][c] = vzero;

#pragma unroll
  for (int t = 0; t < NT; ++t) {
    const float* __restrict__ A  = (t == 0) ? A0  : ((t == 1) ? A1  : A2);
    const float* __restrict__ BT = (t == 0) ? BT0 : ((t == 1) ? BT1 : BT2);

    // fixed per-lane base pointers; all K-step offsets are immediates
    const float* pa0 = A + (size_t)mrow[0] * K + lk;
    const float* pa1 = A + (size_t)mrow[1] * K + lk;
    const float* pa2 = A + (size_t)mrow[2] * K + lk;
    const float* pa3 = A + (size_t)mrow[3] * K + lk;
    const float* pb0 = BT + (size_t)n0 * K + lk;
    const float* pb1 = BT + (size_t)n1 * K + lk;

    // prologue: fetch first K=4 fragment
    v2f a_c[4], b_c[2];
    a_c[0] = *(const v2f*)pa0;
    a_c[1] = *(const v2f*)pa1;
    a_c[2] = *(const v2f*)pa2;
    a_c[3] = *(const v2f*)pa3;
    b_c[0] = *(const v2f*)pb0;
    b_c[1] = *(const v2f*)pb1;

#pragma unroll 4
    for (int k = 4; k < K; k += 4) {
      // prefetch next fragment (overlaps with WMMAs below)
      v2f a_n[4], b_n[2];
      a_n[0] = *(const v2f*)(pa0 + k);
      a_n[1] = *(const v2f*)(pa1 + k);
      a_n[2] = *(const v2f*)(pa2 + k);
      a_n[3] = *(const v2f*)(pa3 + k);
      b_n[0] = *(const v2f*)(pb0 + k);
      b_n[1] = *(const v2f*)(pb1 + k);

#pragma unroll
      for (int r = 0; r < 4; ++r)
#pragma unroll
        for (int c = 0; c < 2; ++c)
          acc[r][c] = __builtin_amdgcn_wmma_f32_16x16x4_f32(
              false, a_c[r], false, b_c[c], (short)0, acc[r][c], false, false);

#pragma unroll
      for (int r = 0; r < 4; ++r) a_c[r] = a_n[r];
      b_c[0] = b_n[0]; b_c[1] = b_n[1];
    }

    // epilogue fragment
#pragma unroll
    for (int r = 0; r < 4; ++r)
#pragma unroll
      for (int c = 0; c < 2; ++c)
        acc[r][c] = __builtin_amdgcn_wmma_f32_16x16x4_f32(
            false, a_c[r], false, b_c[c], (short)0, acc[r][c], false, false);
  }

  // C/D layout: lane 0-15 -> M=v, lane 16-31 -> M=v+8; N = lane&15
  const int mOff = (lane < 16) ? 0 : 8;
#pragma unroll
  for (int c = 0; c < 2; ++c) {
    const int n = (c == 0) ? n0v : n1v;
    if (n >= M) continue;
    const float bv = bias[n];
#pragma unroll
    for (int r = 0; r < 4; ++r) {
      if (!rowOK[r]) continue;
#pragma unroll
      for (int v = 0; v < 8; ++v) {
        const int m = rowBase + r * 16 + mOff + v;
        C[(size_t)m * LDC + n] = acc[r][c][v] + bv;
      }
    }
  }
}

// ---------------- batch norm over nodes ----------------
__global__ void bn_partial(const float* __restrict__ H, float* __restrict__ sum,
                           float* __restrict__ sumsq, int Nrows, int F) {
  int c = blockIdx.x * blockDim.x + threadIdx.x;
  if (c >= F) return;
  float s = 0.f, q = 0.f;
  for (int r = blockIdx.y; r < Nrows; r += gridDim.y) {
    float v = H[(size_t)r * F + c];
    s += v; q += v * v;
  }
  atomicAdd(&sum[c], s);
  atomicAdd(&sumsq[c], q);
}

__global__ void bn_finalize(float* __restrict__ sum, float* __restrict__ sumsq,
                            const float* __restrict__ g, const float* __restrict__ beta,
                            int Nrows, int F) {
  int c = blockIdx.x * blockDim.x + threadIdx.x;
  if (c >= F) return;
  float inv_n = 1.f / (float)Nrows;
  float mu  = sum[c] * inv_n;
  float var = sumsq[c] * inv_n - mu * mu;
  float a = g[c] * rsqrtf(var + 1e-5f);
  sum[c]   = a;                 // scale
  sumsq[c] = beta[c] - mu * a;  // shift
}

__global__ void bn_apply(float* __restrict__ H, const float* __restrict__ a,
                         const float* __restrict__ b, long n, int F, int relu) {
  long i = (long)blockIdx.x * blockDim.x + threadIdx.x;
  if (i >= n) return;
  int c = (int)(i % F);
  float v = H[i] * a[c] + b[c];
  if (relu) v = fmaxf(v, 0.f);
  H[i] = v;
}

// ---------------- linear head + log_softmax (one wave per row) ----------------
__global__ __launch_bounds__(256)
void head_kernel(const float* __restrict__ H, const float* __restrict__ Wl,
                 const float* __restrict__ bl, float* __restrict__ out, int Nrows) {
  const int gwave = (blockIdx.x * blockDim.x + threadIdx.x) >> 5;
  const int lane = threadIdx.x & 31;
  if (gwave >= Nrows) return;
  const float* __restrict__ hrow = H + (size_t)gwave * 300;
  float acc[10];
#pragma unroll
  for (int j = 0; j < 10; ++j) acc[j] = 0.f;
  for (int k = lane; k < 300; k += 32) {
    const float hv = hrow[k];
#pragma unroll
    for (int j = 0; j < 10; ++j) acc[j] += hv * Wl[k * 10 + j];
  }
#pragma unroll
  for (int j = 0; j < 10; ++j) {
#pragma unroll
    for (int off = 16; off >= 1; off >>= 1)
      acc[j] += __shfl_xor(acc[j], off, 32);
  }
  if (lane == 0) {
    float l[10], m = -1e30f;
#pragma unroll
    for (int j = 0; j < 10; ++j) { l[j] = acc[j] + bl[j]; m = fmaxf(m, l[j]); }
    float s = 0.f;
#pragma unroll
    for (int j = 0; j < 10; ++j) s += expf(l[j] - m);
    const float lse = logf(s);
#pragma unroll
    for (int j = 0; j < 10; ++j) out[(size_t)gwave * 10 + j] = l[j] - m - lse;
  }
}

// ---------------- orchestration ----------------
extern "C" void kernel_launch(void* const* d_in, const int* in_sizes, int n_in,
                              void* d_out, int out_size, void* d_ws, size_t ws_size,
                              hipStream_t stream) {
  (void)in_sizes; (void)n_in; (void)out_size; (void)ws_size;
  const float* x   = (const float*)d_in[0];
  const int*   ei  = (const int*)d_in[1];
  const float* ew  = (const float*)d_in[2];
  const float* W1a = (const float*)d_in[3];
  const float* b1a = (const float*)d_in[4];
  const float* W1b = (const float*)d_in[5];
  const float* b1b = (const float*)d_in[6];
  const float* W1c = (const float*)d_in[7];
  const float* b1c = (const float*)d_in[8];
  const float* g1  = (const float*)d_in[9];
  const float* be1 = (const float*)d_in[10];
  const float* W2a = (const float*)d_in[11];
  const float* b2a = (const float*)d_in[12];
  const float* W2b = (const float*)d_in[13];
  const float* b2b = (const float*)d_in[14];
  const float* W2c = (const float*)d_in[15];
  const float* b2c = (const float*)d_in[16];
  const float* g2  = (const float*)d_in[17];
  const float* be2 = (const float*)d_in[18];
  const float* Wl  = (const float*)d_in[19];
  const float* bl  = (const float*)d_in[20];
  float* out = (float*)d_out;

  const int N = N_NODES, E = N_EDGES;
  const int* row = ei;
  const int* col = ei + E;

  constexpr size_t S1 = (size_t)512 * 400;   // one layer-1 weight matrix
  constexpr size_t S2 = (size_t)1200 * 100;  // one layer-2 weight matrix

  float* ws = (float*)d_ws;
  size_t off = 0;
  float* deg  = ws + off; off += N;
  float* norm = ws + off; off += E;
  float* stA  = ws + off; off += 1200;   // sums -> BN scale
  float* stB  = ws + off; off += 1200;   // sumsq -> BN shift
  off = (off + 255) & ~(size_t)255;
  float* wt1 = ws + off; off += 6 * S1;  // transposed layer-1 weights [M x K]
  float* wt2 = ws + off; off += 6 * S2;  // transposed layer-2 weights [M x K]
  off = (off + 255) & ~(size_t)255;
  float* T1 = ws + off; off += (size_t)N * 1200;   // T1_1 (512) then T1_2 (1200)
  float* T2 = ws + off; off += (size_t)N * 1200;   // T2_1 (512) then T2_2 (1200)
  float* H1 = ws + off; off += (size_t)N * 1200;
  float* H2 = ws + off; off += (size_t)N * 300;

  // ---- weight transposes (tiny) ----
  {
    const int n1 = 512 * 400, n2 = 1200 * 100;
    transpose_w<<<cdiv(n1, 256), 256, 0, stream>>>(W1a,          wt1 + 0 * S1, 512, 400);
    transpose_w<<<cdiv(n1, 256), 256, 0, stream>>>(W1b,          wt1 + 1 * S1, 512, 400);
    transpose_w<<<cdiv(n1, 256), 256, 0, stream>>>(W1b + S1,     wt1 + 2 * S1, 512, 400);
    transpose_w<<<cdiv(n1, 256), 256, 0, stream>>>(W1c,          wt1 + 3 * S1, 512, 400);
    transpose_w<<<cdiv(n1, 256), 256, 0, stream>>>(W1c + S1,     wt1 + 4 * S1, 512, 400);
    transpose_w<<<cdiv(n1, 256), 256, 0, stream>>>(W1c + 2 * S1, wt1 + 5 * S1, 512, 400);
    transpose_w<<<cdiv(n2, 256), 256, 0, stream>>>(W2a,          wt2 + 0 * S2, 1200, 100);
    transpose_w<<<cdiv(n2, 256), 256, 0, stream>>>(W2b,          wt2 + 1 * S2, 1200, 100);
    transpose_w<<<cdiv(n2, 256), 256, 0, stream>>>(W2b + S2,     wt2 + 2 * S2, 1200, 100);
    transpose_w<<<cdiv(n2, 256), 256, 0, stream>>>(W2c,          wt2 + 3 * S2, 1200, 100);
    transpose_w<<<cdiv(n2, 256), 256, 0, stream>>>(W2c + S2,     wt2 + 4 * S2, 1200, 100);
    transpose_w<<<cdiv(n2, 256), 256, 0, stream>>>(W2c + 2 * S2, wt2 + 5 * S2, 1200, 100);
  }

  // ---- edge norm ----
  hipMemsetAsync(deg, 0, (size_t)N * sizeof(float), stream);
  deg_kernel<<<cdiv(E, 256), 256, 0, stream>>>(row, ew, deg, E);
  dis_kernel<<<cdiv(N, 256), 256, 0, stream>>>(deg, N);
  norm_kernel<<<cdiv(E, 256), 256, 0, stream>>>(row, col, ew, deg, norm, E);

  // ---- layer-1 Chebyshev propagations (D=512) ----
  hipMemsetAsync(T1, 0, (size_t)N * 512 * sizeof(float), stream);
  prop_scatter<<<E, 128, 0, stream>>>(x, row, col, norm, T1, 512);
  hipMemsetAsync(T2, 0, (size_t)N * 512 * sizeof(float), stream);
  prop_scatter<<<E, 128, 0, stream>>>(T1, row, col, norm, T2, 512);
  cheb2_combine<<<cdiv((long)N * 512, 256), 256, 0, stream>>>(T2, x, (long)N * 512);

  // ---- layer-1 GEMMs (K=512, chunks of 400 into H1, ldc=1200) ----
  {
    dim3 grid(cdiv(N, 256), cdiv(400, 64));
    gemm_cheb<1, 512, 400, 1200><<<grid, 256, 0, stream>>>(
        x, x, x, wt1 + 0 * S1, wt1 + 0 * S1, wt1 + 0 * S1, N, b1a, H1 + 0);
    gemm_cheb<2, 512, 400, 1200><<<grid, 256, 0, stream>>>(
        x, T1, T1, wt1 + 1 * S1, wt1 + 2 * S1, wt1 + 2 * S1, N, b1b, H1 + 400);
    gemm_cheb<3, 512, 400, 1200><<<grid, 256, 0, stream>>>(
        x, T1, T2, wt1 + 3 * S1, wt1 + 4 * S1, wt1 + 5 * S1, N, b1c, H1 + 800);
  }

  // ---- BN1 + ReLU on H1 (F=1200) ----
  hipMemsetAsync(stA, 0, 1200 * sizeof(float), stream);
  hipMemsetAsync(stB, 0, 1200 * sizeof(float), stream);
  {
    dim3 bgrid(cdiv(1200, 256), 256);
    bn_partial<<<bgrid, 256, 0, stream>>>(H1, stA, stB, N, 1200);
    bn_finalize<<<cdiv(1200, 256), 256, 0, stream>>>(stA, stB, g1, be1, N, 1200);
    bn_apply<<<cdiv((long)N * 1200, 256), 256, 0, stream>>>(H1, stA, stB, (long)N * 1200, 1200, 1);
  }

  // ---- layer-2 Chebyshev propagations (D=1200) ----
  hipMemsetAsync(T1, 0, (size_t)N * 1200 * sizeof(float), stream);
  prop_scatter<<<E, 128, 0, stream>>>(H1, row, col, norm, T1, 1200);
  hipMemsetAsync(T2, 0, (size_t)N * 1200 * sizeof(float), stream);
  prop_scatter<<<E, 128, 0, stream>>>(T1, row, col, norm, T2, 1200);
  cheb2_combine<<<cdiv((long)N * 1200, 256), 256, 0, stream>>>(T2, H1, (long)N * 1200);

  // ---- layer-2 GEMMs (K=1200, chunks of 100 into H2, ldc=300) ----
  {
    dim3 grid(cdiv(N, 256), cdiv(100, 64));
    gemm_cheb<1, 1200, 100, 300><<<grid, 256, 0, stream>>>(
        H1, H1, H1, wt2 + 0 * S2, wt2 + 0 * S2, wt2 + 0 * S2, N, b2a, H2 + 0);
    gemm_cheb<2, 1200, 100, 300><<<grid, 256, 0, stream>>>(
        H1, T1, T1, wt2 + 1 * S2, wt2 + 2 * S2, wt2 + 2 * S2, N, b2b, H2 + 100);
    gemm_cheb<3, 1200, 100, 300><<<grid, 256, 0, stream>>>(
        H1, T1, T2, wt2 + 3 * S2, wt2 + 4 * S2, wt2 + 5 * S2, N, b2c, H2 + 200);
  }

  // ---- BN2 on H2 (F=300, no ReLU) ----
  hipMemsetAsync(stA, 0, 300 * sizeof(float), stream);
  hipMemsetAsync(stB, 0, 300 * sizeof(float), stream);
  {
    dim3 bgrid(cdiv(300, 256), 256);
    bn_partial<<<bgrid, 256, 0, stream>>>(H2, stA, stB, N, 300);
    bn_finalize<<<cdiv(300, 256), 256, 0, stream>>>(stA, stB, g2, be2, N, 300);
    bn_apply<<<cdiv((long)N * 300, 256), 256, 0, stream>>>(H2, stA, stB, (long)N * 300, 300, 0);
  }

  // ---- linear head + log_softmax ----
  head_kernel<<<cdiv((long)N * 32, 256), 256, 0, stream>>>(H2, Wl, bl, out, N);
}